// GPUChorus_8323646620201
// MI455X (gfx1250) — compile-verified
//
#include <hip/hip_runtime.h>

#define SR      16000
#define T_LEN   480000
#define B_N     64
#define TILE    8192
#define HALO    800           // MAX_DELAY_SAMPLES
#define MAXDF   800.0f

typedef unsigned int v4u __attribute__((ext_vector_type(4)));
typedef int          v8i __attribute__((ext_vector_type(8)));
typedef int          v4i __attribute__((ext_vector_type(4)));
typedef float        v4f __attribute__((ext_vector_type(4)));

__global__ __launch_bounds__(256) void GPUChorus_kernel(
    const float* __restrict__ audio,
    const float* __restrict__ rate_hz,
    const float* __restrict__ depth,
    const float* __restrict__ centre_ms,
    const float* __restrict__ mix,
    float* __restrict__ out)
{
    __shared__ float win[TILE + HALO];

    const int b        = blockIdx.y;
    const int s        = blockIdx.x * TILE;
    const int win_base = s - HALO;                         // global idx mapped to local 0
    const int src_start = (win_base < 0) ? 0 : win_base;   // clamp at row start
    const int lds_off   = src_start - win_base;            // 800 for tile 0, else 0
    const int tile_len  = (T_LEN - s < TILE) ? (T_LEN - s) : TILE;  // multiple of 4
    const int win_len   = (s + tile_len) - src_start;      // elements to DMA

    // ---- Tensor Data Mover: DMA window of row b into LDS (wave 0 only) ----
    if (threadIdx.x < 32) {
        unsigned long long gaddr =
            (unsigned long long)(const void*)(audio + (size_t)b * T_LEN + src_start);
        unsigned lds_addr = (unsigned)(unsigned long long)(const void*)(win + lds_off);

        v4u g0;
        g0.x = 1u;                                          // count=1, user descriptor
        g0.y = lds_addr;                                    // LDS byte address
        g0.z = (unsigned)(gaddr & 0xFFFFFFFFull);           // global addr [31:0]
        g0.w = (unsigned)((gaddr >> 32) & 0x01FFFFFFull)    // global addr [56:32]
             | (2u << 30);                                  // type = 2 ("image")

        const unsigned td0   = (unsigned)win_len;           // tensor_dim0 (32b)
        const unsigned td1   = 1u;                          // tensor_dim1
        const unsigned tile0 = (unsigned)win_len;           // tile_dim0 (16b, <= 8992)

        v8i g1;
        g1[0] = (int)(2u << 16);                            // data_size=2 -> 4 bytes
        g1[1] = (int)((td0 & 0xFFFFu) << 16);               // tensor_dim0[15:0] @ bit48
        g1[2] = (int)((td0 >> 16) | ((td1 & 0xFFFFu) << 16));
        g1[3] = (int)((td1 >> 16) | (tile0 << 16));         // tile_dim0 @ bit112
        g1[4] = (int)1;                                     // tile_dim1=1, tile_dim2=0
        g1[5] = (int)td0;                                   // tensor_dim0_stride lo
        g1[6] = 0;
        g1[7] = 0;

        v4i gz = {0, 0, 0, 0};
#if defined(__clang_major__) && (__clang_major__ >= 23)
        v8i gz8 = {0, 0, 0, 0, 0, 0, 0, 0};
        __builtin_amdgcn_tensor_load_to_lds(g0, g1, gz, gz, gz8, 0);
#else
        __builtin_amdgcn_tensor_load_to_lds(g0, g1, gz, gz, 0);
#endif
        __builtin_amdgcn_s_wait_tensorcnt(0);
    }
    __syncthreads();

    // ---- per-sample chorus math, all taps served from LDS ----
    const float rate = rate_hz[b];
    const float dep  = depth[b];
    const float cds  = centre_ms[b] * (SR / 1000.0f);
    const float dcds = dep * cds;
    const float mx   = mix[b];
    const float omix = 1.0f - mx;
    const float wrad = 6.28318530717958647692f * rate * (1.0f / (float)SR);

    float* orow = out + (size_t)b * T_LEN;
    const int nvec = tile_len >> 2;                // tile_len is a multiple of 4

    for (int v = threadIdx.x; v < nvec; v += 256) {
        const int i0 = v << 2;                     // local sample index of lane's quad
        v4f r;
#pragma unroll
        for (int k = 0; k < 4; ++k) {
            const int   i   = i0 + k;
            const int   pos = s + i;
            const float lfo = __sinf(wrad * (float)pos);
            float delay = cds + lfo * dcds;
            delay = fminf(fmaxf(delay, 1.0f), MAXDF);

            const float rp   = (float)pos - delay;
            const float lowf = floorf(rp);
            const float frac = rp - lowf;

            int il = (int)fminf(fmaxf(lowf, 0.0f), (float)(T_LEN - 1));
            int ih = il + 1;
            if (ih > T_LEN - 1) ih = T_LEN - 1;

            const float al = win[il - win_base];   // LDS gather (always in DMA'd region)
            const float ah = win[ih - win_base];
            float delayed = al * (1.0f - frac) + ah * frac;
            delayed = (rp >= 0.0f) ? delayed : 0.0f;

            const float dry = win[HALO + i];       // pos - win_base == HALO + i
            r[k] = dry * omix + delayed * mx;
        }
        // one 128-bit non-temporal store per lane: pure write stream, keep it out of L2
        __builtin_nontemporal_store(r, (v4f*)(orow + s + i0));
    }
}

extern "C" void kernel_launch(void* const* d_in, const int* in_sizes, int n_in,
                              void* d_out, int out_size, void* d_ws, size_t ws_size,
                              hipStream_t stream) {
    const float* audio     = (const float*)d_in[0];
    const float* rate_hz   = (const float*)d_in[1];
    const float* depth     = (const float*)d_in[2];
    const float* centre_ms = (const float*)d_in[3];
    // d_in[4] = feedback: unused by the reference math
    const float* mix       = (const float*)d_in[5];
    float* out = (float*)d_out;

    dim3 grid((T_LEN + TILE - 1) / TILE, B_N);
    GPUChorus_kernel<<<grid, 256, 0, stream>>>(audio, rate_hz, depth, centre_ms, mix, out);
}